// TimeSwiGLU_36893769073474
// MI455X (gfx1250) — compile-verified
//
#include <hip/hip_runtime.h>

// TimeSwiGLU-style chained causal attention (no softmax), fp32, via
// V_WMMA_F32_16X16X4_F32 on gfx1250 (wave32).
//
// Pass 1: t12[b,h,s,d] = silu((tril(Q1 K1^T) V)) * (tril(Q2 K2^T) V)
// Pass 2: out[b,t,h,d] = tril(Q3 K3^T) @ t12
//
// qkv layout [B,S,7,H,D], i = {q1,k1,q2,k2,q3,k3,v}.

typedef __attribute__((ext_vector_type(2))) float v2f;
typedef __attribute__((ext_vector_type(4))) float v4f;
typedef __attribute__((ext_vector_type(8))) float v8f;

namespace {

constexpr int H  = 8;
constexpr int D  = 64;
constexpr int NI = 7;

__device__ __forceinline__ v8f wmma_f32(v2f a, v2f b, v8f c) {
  // D = A(16x4,f32) * B(4x16,f32) + C(16x16,f32)
  return __builtin_amdgcn_wmma_f32_16x16x4_f32(
      /*neg_a=*/false, a, /*neg_b=*/false, b,
      /*c_mod=*/(short)0, c, /*reuse_a=*/false, /*reuse_b=*/false);
}

} // namespace

// ---------------------------------------------------------------------------
// Pass 1: per wave = one 16-row query tile; accumulate t1 = a1@V, t2 = a2@V.
// ---------------------------------------------------------------------------
__global__ __launch_bounds__(128) void
tswiglu_pass1(const float* __restrict__ qkv, float* __restrict__ t12,
              int B, int S) {
  const int b      = blockIdx.x / H;
  const int h      = blockIdx.x % H;
  const int base_t = blockIdx.y * 64;
  const int wave   = threadIdx.x >> 5;
  const int lane   = threadIdx.x & 31;
  const int half   = lane >> 4;      // which 16-lane half
  const int lr     = lane & 15;      // row/col within 16
  const int kb     = half * 2;       // K-pair base for A/B fragments
  const int t0     = base_t + wave * 16;

  __shared__ float lk1[16 * 64];
  __shared__ float lk2[16 * 64];
  __shared__ float lv [16 * 64];
  __shared__ float lsc[4][16 * 17];  // per-wave score scratch, stride 17

  auto row = [&](int s, int i) -> const float* {
    return qkv + (((long)b * S + s) * NI + i) * (long)(H * D) + (long)h * D;
  };

  // Preload Q1/Q2 A-fragments for this wave's 16 query rows (kept in VGPRs).
  v2f q1f[16], q2f[16];
  {
    const float* r1 = row(t0 + lr, 0);
    const float* r2 = row(t0 + lr, 2);
#pragma unroll
    for (int k4 = 0; k4 < 16; ++k4) {
      q1f[k4].x = r1[k4 * 4 + kb];
      q1f[k4].y = r1[k4 * 4 + kb + 1];
      q2f[k4].x = r2[k4 * 4 + kb];
      q2f[k4].y = r2[k4 * 4 + kb + 1];
    }
  }

  const v8f vzero = {};
  v8f t1a[4], t2a[4];
#pragma unroll
  for (int j = 0; j < 4; ++j) { t1a[j] = vzero; t2a[j] = vzero; }

  const int sEnd = base_t + 48;                 // max t0 among the 4 waves
  for (int s0 = 0; s0 <= sEnd; s0 += 16) {
    __syncthreads();                            // previous tile fully consumed
    // Cooperative coalesced staging of K1, K2, V key tiles (16x64 fp32 each).
#pragma unroll
    for (int it = 0; it < 2; ++it) {
      const int idx = threadIdx.x + it * 128;   // 256 float4 slots per tile
      const int r = idx >> 4, c = (idx & 15) * 4;
      *(v4f*)&lk1[r * 64 + c] = *(const v4f*)(row(s0 + r, 1) + c);
      *(v4f*)&lk2[r * 64 + c] = *(const v4f*)(row(s0 + r, 3) + c);
      *(v4f*)&lv [r * 64 + c] = *(const v4f*)(row(s0 + r, 6) + c);
    }
    __syncthreads();

    if (s0 > t0) continue;                      // causal: wave has no work here
    const bool diag = (s0 == t0);
    float* sc = lsc[wave];

    // -------- a1 = Q1 K1^T tile, then t1a += a1 @ V --------
    {
      v8f s1 = vzero;
#pragma unroll
      for (int k4 = 0; k4 < 16; ++k4) {         // contract D=64 in K=4 steps
        v2f bf;                                  // B[d][s] = K[s][d]
        bf.x = lk1[lr * 64 + k4 * 4 + kb];
        bf.y = lk1[lr * 64 + k4 * 4 + kb + 1];
        s1 = wmma_f32(q1f[k4], bf, s1);
      }
#pragma unroll
      for (int i = 0; i < 8; ++i) {              // mask + C-layout -> LDS
        const int m = i + 8 * half;
        float v = s1[i];
        if (diag && lr > m) v = 0.0f;            // tril
        sc[m * 17 + lr] = v;
      }
      asm volatile("s_wait_dscnt 0" ::: "memory");
#pragma unroll
      for (int k4 = 0; k4 < 4; ++k4) {           // contract S-tile (K=16)
        v2f af;                                  // A-layout reload of scores
        af.x = sc[lr * 17 + k4 * 4 + kb];
        af.y = sc[lr * 17 + k4 * 4 + kb + 1];
#pragma unroll
        for (int j = 0; j < 4; ++j) {            // 4 N-tiles cover D=64
          v2f bf;
          bf.x = lv[(k4 * 4 + kb    ) * 64 + j * 16 + lr];
          bf.y = lv[(k4 * 4 + kb + 1) * 64 + j * 16 + lr];
          t1a[j] = wmma_f32(af, bf, t1a[j]);
        }
      }
    }

    // -------- a2 = Q2 K2^T tile, then t2a += a2 @ V --------
    {
      v8f s2 = vzero;
#pragma unroll
      for (int k4 = 0; k4 < 16; ++k4) {
        v2f bf;
        bf.x = lk2[lr * 64 + k4 * 4 + kb];
        bf.y = lk2[lr * 64 + k4 * 4 + kb + 1];
        s2 = wmma_f32(q2f[k4], bf, s2);
      }
#pragma unroll
      for (int i = 0; i < 8; ++i) {
        const int m = i + 8 * half;
        float v = s2[i];
        if (diag && lr > m) v = 0.0f;
        sc[m * 17 + lr] = v;
      }
      asm volatile("s_wait_dscnt 0" ::: "memory");
#pragma unroll
      for (int k4 = 0; k4 < 4; ++k4) {
        v2f af;
        af.x = sc[lr * 17 + k4 * 4 + kb];
        af.y = sc[lr * 17 + k4 * 4 + kb + 1];
#pragma unroll
        for (int j = 0; j < 4; ++j) {
          v2f bf;
          bf.x = lv[(k4 * 4 + kb    ) * 64 + j * 16 + lr];
          bf.y = lv[(k4 * 4 + kb + 1) * 64 + j * 16 + lr];
          t2a[j] = wmma_f32(af, bf, t2a[j]);
        }
      }
    }
  }

  // Epilogue: t12[b,h,t,d] = silu(t1) * t2  (contiguous [B,H,S,D] for pass 2)
#pragma unroll
  for (int j = 0; j < 4; ++j) {
#pragma unroll
    for (int i = 0; i < 8; ++i) {
      const int m  = i + 8 * half;
      const float x = t1a[j][i];
      const float sig = 1.0f / (1.0f + __expf(-x));
      t12[(((long)b * H + h) * S + (t0 + m)) * D + j * 16 + lr] =
          (x * sig) * t2a[j][i];
    }
  }
}

// ---------------------------------------------------------------------------
// Pass 2: out = tril(Q3 K3^T) @ t12
// ---------------------------------------------------------------------------
__global__ __launch_bounds__(128) void
tswiglu_pass2(const float* __restrict__ qkv, const float* __restrict__ t12,
              float* __restrict__ out, int B, int S) {
  const int b      = blockIdx.x / H;
  const int h      = blockIdx.x % H;
  const int base_t = blockIdx.y * 64;
  const int wave   = threadIdx.x >> 5;
  const int lane   = threadIdx.x & 31;
  const int half   = lane >> 4;
  const int lr     = lane & 15;
  const int kb     = half * 2;
  const int t0     = base_t + wave * 16;

  __shared__ float lk3[16 * 64];
  __shared__ float lvt[16 * 64];
  __shared__ float lsc[4][16 * 17];

  auto row = [&](int s, int i) -> const float* {
    return qkv + (((long)b * S + s) * NI + i) * (long)(H * D) + (long)h * D;
  };
  const float* vbase = t12 + (((long)b * H + h) * S) * D;

  v2f q3f[16];
  {
    const float* r3 = row(t0 + lr, 4);
#pragma unroll
    for (int k4 = 0; k4 < 16; ++k4) {
      q3f[k4].x = r3[k4 * 4 + kb];
      q3f[k4].y = r3[k4 * 4 + kb + 1];
    }
  }

  const v8f vzero = {};
  v8f oa[4];
#pragma unroll
  for (int j = 0; j < 4; ++j) oa[j] = vzero;

  const int sEnd = base_t + 48;
  for (int s0 = 0; s0 <= sEnd; s0 += 16) {
    __syncthreads();
#pragma unroll
    for (int it = 0; it < 2; ++it) {
      const int idx = threadIdx.x + it * 128;
      const int r = idx >> 4, c = (idx & 15) * 4;
      *(v4f*)&lk3[r * 64 + c] = *(const v4f*)(row(s0 + r, 5) + c);
      *(v4f*)&lvt[r * 64 + c] = *(const v4f*)(vbase + (long)(s0 + r) * D + c);
    }
    __syncthreads();

    if (s0 > t0) continue;
    const bool diag = (s0 == t0);
    float* sc = lsc[wave];

    v8f s3 = vzero;
#pragma unroll
    for (int k4 = 0; k4 < 16; ++k4) {
      v2f bf;
      bf.x = lk3[lr * 64 + k4 * 4 + kb];
      bf.y = lk3[lr * 64 + k4 * 4 + kb + 1];
      s3 = wmma_f32(q3f[k4], bf, s3);
    }
#pragma unroll
    for (int i = 0; i < 8; ++i) {
      const int m = i + 8 * half;
      float v = s3[i];
      if (diag && lr > m) v = 0.0f;
      sc[m * 17 + lr] = v;
    }
    asm volatile("s_wait_dscnt 0" ::: "memory");
#pragma unroll
    for (int k4 = 0; k4 < 4; ++k4) {
      v2f af;
      af.x = sc[lr * 17 + k4 * 4 + kb];
      af.y = sc[lr * 17 + k4 * 4 + kb + 1];
#pragma unroll
      for (int j = 0; j < 4; ++j) {
        v2f bf;
        bf.x = lvt[(k4 * 4 + kb    ) * 64 + j * 16 + lr];
        bf.y = lvt[(k4 * 4 + kb + 1) * 64 + j * 16 + lr];
        oa[j] = wmma_f32(af, bf, oa[j]);
      }
    }
  }

  // out layout [B,S,H,D]
#pragma unroll
  for (int j = 0; j < 4; ++j) {
#pragma unroll
    for (int i = 0; i < 8; ++i) {
      const int m = i + 8 * half;
      out[(((long)b * S + (t0 + m)) * H + h) * D + j * 16 + lr] = oa[j][i];
    }
  }
}

extern "C" void kernel_launch(void* const* d_in, const int* in_sizes, int n_in,
                              void* d_out, int out_size, void* d_ws, size_t ws_size,
                              hipStream_t stream) {
  const float* qkv = (const float*)d_in[0];
  const int S = 2048;
  const int B = (int)((long)in_sizes[0] / ((long)S * NI * H * D));
  float* t12 = (float*)d_ws;                 // B*H*S*D fp32 = 8.4 MB scratch
  float* outp = (float*)d_out;

  dim3 grid(B * H, S / 64);
  dim3 block(128);
  tswiglu_pass1<<<grid, block, 0, stream>>>(qkv, t12, B, S);
  tswiglu_pass2<<<grid, block, 0, stream>>>(qkv, t12, outp, B, S);
}